// HexPlaneStaticResidualField_22514218566154
// MI455X (gfx1250) — compile-verified
//
#include <hip/hip_runtime.h>
#include <cstdint>

typedef float v2f __attribute__((ext_vector_type(2)));
typedef float v8f __attribute__((ext_vector_type(8)));

#define NLEVELS 3
#define ODIM 32
#define TRES 100

struct Params {
  const float* pts;
  const float* ts;
  const float* tw;
  const float* sg[NLEVELS][3];   // transposed [H][W][32] static planes
  const float* rg[NLEVELS][3];   // transposed [H][W][32] residual planes
  const float* sW[NLEVELS];
  const float* sb[NLEVELS];
  const float* rW[NLEVELS];
  const float* rb[NLEVELS];
  float* out;
  int n_groups;
  int npts;
};

// ---------------------------------------------------------------------------
// Layout transform: [32,H,W] -> [H,W,32] so a texel's 32 channels are one
// contiguous 128B read (lanes-over-channels gather hits 2 cachelines, not 32).
// ---------------------------------------------------------------------------
__global__ void transpose_chw_hwc(const float* __restrict__ src,
                                  float* __restrict__ dst, int HW) {
  int i = blockIdx.x * 256 + threadIdx.x;
  if (i < HW * ODIM) {
    int c  = i & (ODIM - 1);
    int hw = i >> 5;
    dst[i] = src[(size_t)c * HW + hw];   // coalesced writes, L2-hit reads
  }
}

// Bilinear sample from transposed grid [H][W][32], channel c.
__device__ __forceinline__ float bsample(const float* __restrict__ g,
                                         float cx, float cy, int W, int H, int c) {
  float fx = (cx + 1.0f) * 0.5f * (float)(W - 1);
  float fy = (cy + 1.0f) * 0.5f * (float)(H - 1);
  fx = fminf(fmaxf(fx, 0.0f), (float)(W - 1));
  fy = fminf(fmaxf(fy, 0.0f), (float)(H - 1));
  float x0f = floorf(fx), y0f = floorf(fy);
  int x0 = (int)x0f, y0 = (int)y0f;
  int x1 = min(x0 + 1, W - 1), y1 = min(y0 + 1, H - 1);
  float wx = fx - x0f, wy = fy - y0f;
  const float* r0 = g + (size_t)y0 * W * ODIM;
  const float* r1 = g + (size_t)y1 * W * ODIM;
  float v00 = r0[x0 * ODIM + c], v01 = r0[x1 * ODIM + c];
  float v10 = r1[x0 * ODIM + c], v11 = r1[x1 * ODIM + c];
  float top = v00 + wx * (v01 - v00);
  float bot = v10 + wx * (v11 - v10);
  return top + wy * (bot - top);
}

// ---------------------------------------------------------------------------
// Main kernel: one wave per group of 16 points. Features staged in LDS
// (lane = channel), 32x32 matmuls via V_WMMA_F32_16X16X4_F32.
// ---------------------------------------------------------------------------
__launch_bounds__(256, 2)
__global__ void hexplane_kernel(Params P) {
  __shared__ float s_coords[8][16][4];
  __shared__ float s_aw[8][16];
  __shared__ __align__(16) float s_feat[8][16][36];   // stride 36: bank-conflict-free

  const int lane = threadIdx.x & 31;
  const int wid  = threadIdx.x >> 5;
  int g = blockIdx.x * 8 + wid;
  const bool active = (g < P.n_groups);
  if (!active) g = 0;                    // uniform clamp; keep EXEC full for WMMA
  const int base = g * 16;

  // ---- per-point coords + time weight (lanes 0..15) ----
  if (lane < 16) {
    int p = min(base + lane, P.npts - 1);
    float x = P.pts[3 * p + 0], y = P.pts[3 * p + 1], z = P.pts[3 * p + 2];
    const float inv = -1.0f / 1.3f;                  // p = -pts / BOUNDS
    float t  = P.ts[p];
    float tc = fminf(fmaxf(t, 0.0f), 1.0f);
    int tidx = min(max((int)(tc * 99.0f), 0), TRES - 1);
    float w  = P.tw[tidx];
    s_coords[wid][lane][0] = x * inv;
    s_coords[wid][lane][1] = y * inv;
    s_coords[wid][lane][2] = z * inv;
    s_coords[wid][lane][3] = tc * 2.0f - 1.0f;
    s_aw[wid][lane] = 1.0f / (1.0f + __expf(-w));    // sigmoid
  }
  __syncthreads();

  const int m = lane & 15;   // WMMA: A-row / B-col / D-col index
  const int h = lane >> 4;   // K-half select

  #pragma unroll
  for (int l = 0; l < NLEVELS; ++l) {
    const int S = 64 << l;   // 64, 128, 256

    // ---- static plane features: f[p][c] = Sxy*Sxz*Syz ----
    #pragma unroll 2
    for (int p = 0; p < 16; ++p) {
      float cx = s_coords[wid][p][0];
      float cy = s_coords[wid][p][1];
      float cz = s_coords[wid][p][2];
      float f = bsample(P.sg[l][0], cx, cy, S, S, lane)
              * bsample(P.sg[l][1], cx, cz, S, S, lane)
              * bsample(P.sg[l][2], cy, cz, S, S, lane);
      s_feat[wid][p][lane] = f;
    }
    __syncthreads();

    // ---- accumulators seeded with sb[n] + rb[n]*aw[m] ----
    const float* sbv = P.sb[l];
    const float* rbv = P.rb[l];
    float sb0 = sbv[m], sb1 = sbv[m + 16];
    float rb0 = rbv[m], rb1 = rbv[m + 16];
    v8f acc0, acc1;
    #pragma unroll
    for (int r = 0; r < 8; ++r) {
      float awm = s_aw[wid][r + 8 * h];
      acc0[r] = sb0 + rb0 * awm;
      acc1[r] = sb1 + rb1 * awm;
    }

    // ---- static 32x32 matmul: D += A(16xK) x B(Kx16), K chunks of 4 ----
    const float* Ws = P.sW[l];
    #pragma unroll
    for (int c = 0; c < 8; ++c) {
      int k = 4 * c + 2 * h;
      v2f a  = *(const v2f*)&s_feat[wid][m][k];          // A[m][k..k+1]
      v2f b0 = *(const v2f*)&Ws[m * ODIM + k];           // B[k][n]=W[n][k], tile n=m
      v2f b1 = *(const v2f*)&Ws[(m + 16) * ODIM + k];    // tile n=m+16
      acc0 = __builtin_amdgcn_wmma_f32_16x16x4_f32(false, a, false, b0,
                                                   (short)0, acc0, false, false);
      acc1 = __builtin_amdgcn_wmma_f32_16x16x4_f32(false, a, false, b1,
                                                   (short)0, acc1, false, false);
    }
    __syncthreads();

    // ---- residual plane features, pre-scaled by aw[p] ----
    #pragma unroll 2
    for (int p = 0; p < 16; ++p) {
      float cx = s_coords[wid][p][0];
      float cy = s_coords[wid][p][1];
      float cz = s_coords[wid][p][2];
      float ct = s_coords[wid][p][3];
      float aw = s_aw[wid][p];
      float f = bsample(P.rg[l][0], cx, ct, S, TRES, lane)
              * bsample(P.rg[l][1], cy, ct, S, TRES, lane)
              * bsample(P.rg[l][2], cz, ct, S, TRES, lane);
      s_feat[wid][p][lane] = f * aw;     // (rf*aw)@W^T == (rf@W^T)*aw row-scale
    }
    __syncthreads();

    const float* Wr = P.rW[l];
    #pragma unroll
    for (int c = 0; c < 8; ++c) {
      int k = 4 * c + 2 * h;
      v2f a  = *(const v2f*)&s_feat[wid][m][k];
      v2f b0 = *(const v2f*)&Wr[m * ODIM + k];
      v2f b1 = *(const v2f*)&Wr[(m + 16) * ODIM + k];
      acc0 = __builtin_amdgcn_wmma_f32_16x16x4_f32(false, a, false, b0,
                                                   (short)0, acc0, false, false);
      acc1 = __builtin_amdgcn_wmma_f32_16x16x4_f32(false, a, false, b1,
                                                   (short)0, acc1, false, false);
    }
    __syncthreads();

    // ---- store D: out[(base+mm)*96 + 32*l + n] ----
    if (active) {
      #pragma unroll
      for (int r = 0; r < 8; ++r) {
        int mm = r + 8 * h;
        if (base + mm < P.npts) {
          size_t row = (size_t)(base + mm) * 96 + (size_t)l * 32;
          P.out[row + m]      = acc0[r];
          P.out[row + m + 16] = acc1[r];
        }
      }
    }
  }
}

// ---------------------------------------------------------------------------
extern "C" void kernel_launch(void* const* d_in, const int* in_sizes, int n_in,
                              void* d_out, int out_size, void* d_ws, size_t ws_size,
                              hipStream_t stream) {
  const int S_tab[3] = {64, 128, 256};

  const float *pts, *ts, *tw;
  const float *sgrid_in[3][3], *rgrid_in[3][3];
  const float *sW[3], *sb[3], *rW[3], *rb[3];

  if (n_in >= 33) {
    // dict order, nested lists flattened leaf-by-leaf
    pts = (const float*)d_in[0];
    ts  = (const float*)d_in[1];
    int idx = 2;
    for (int l = 0; l < 3; ++l) for (int j = 0; j < 3; ++j) sgrid_in[l][j] = (const float*)d_in[idx++];
    for (int l = 0; l < 3; ++l) for (int j = 0; j < 3; ++j) rgrid_in[l][j] = (const float*)d_in[idx++];
    for (int l = 0; l < 3; ++l) sW[l] = (const float*)d_in[idx++];
    for (int l = 0; l < 3; ++l) sb[l] = (const float*)d_in[idx++];
    for (int l = 0; l < 3; ++l) rW[l] = (const float*)d_in[idx++];
    for (int l = 0; l < 3; ++l) rb[l] = (const float*)d_in[idx++];
    tw = (const float*)d_in[idx];
  } else {
    // fallback: one concatenated buffer per dict entry
    pts = (const float*)d_in[0];
    ts  = (const float*)d_in[1];
    const float* sgb = (const float*)d_in[2];
    size_t off = 0;
    for (int l = 0; l < 3; ++l) {
      size_t sz = (size_t)S_tab[l] * S_tab[l] * ODIM;
      for (int j = 0; j < 3; ++j) { sgrid_in[l][j] = sgb + off; off += sz; }
    }
    const float* rgb = (const float*)d_in[3];
    off = 0;
    for (int l = 0; l < 3; ++l) {
      size_t sz = (size_t)TRES * S_tab[l] * ODIM;
      for (int j = 0; j < 3; ++j) { rgrid_in[l][j] = rgb + off; off += sz; }
    }
    const float* swb = (const float*)d_in[4]; for (int l = 0; l < 3; ++l) sW[l] = swb + l * 1024;
    const float* sbb = (const float*)d_in[5]; for (int l = 0; l < 3; ++l) sb[l] = sbb + l * 32;
    const float* rwb = (const float*)d_in[6]; for (int l = 0; l < 3; ++l) rW[l] = rwb + l * 1024;
    const float* rbb = (const float*)d_in[7]; for (int l = 0; l < 3; ++l) rb[l] = rbb + l * 32;
    tw = (const float*)d_in[8];
  }

  // workspace: transposed [H][W][32] grids (~50.2 MB total)
  float* ws = (float*)d_ws;
  size_t off = 0;
  Params P;
  P.pts = pts; P.ts = ts; P.tw = tw; P.out = (float*)d_out;
  for (int l = 0; l < 3; ++l) {
    int S = S_tab[l];
    for (int j = 0; j < 3; ++j) {
      int HW = S * S;
      float* dst = ws + off; off += (size_t)HW * ODIM;
      transpose_chw_hwc<<<(HW * ODIM + 255) / 256, 256, 0, stream>>>(sgrid_in[l][j], dst, HW);
      P.sg[l][j] = dst;
    }
    for (int j = 0; j < 3; ++j) {
      int HW = TRES * S;
      float* dst = ws + off; off += (size_t)HW * ODIM;
      transpose_chw_hwc<<<(HW * ODIM + 255) / 256, 256, 0, stream>>>(rgrid_in[l][j], dst, HW);
      P.rg[l][j] = dst;
    }
    P.sW[l] = sW[l]; P.sb[l] = sb[l]; P.rW[l] = rW[l]; P.rb[l] = rb[l];
  }

  int npts = in_sizes[1];                 // timestamps: N elements
  P.npts = npts;
  P.n_groups = (npts + 15) / 16;
  int nblocks = (P.n_groups + 7) / 8;     // 8 waves (groups) per 256-thread block
  hexplane_kernel<<<nblocks, 256, 0, stream>>>(P);

  (void)n_in; (void)out_size; (void)ws_size;
}